// GATLSTMCell_3599182594880
// MI455X (gfx1250) — compile-verified
//
#include <hip/hip_runtime.h>

typedef float v2f __attribute__((ext_vector_type(2)));
typedef float v8f __attribute__((ext_vector_type(8)));

// Problem constants from the reference
constexpr int B   = 2;
constexpr int N   = 20000;
constexpr int E   = 320000;
constexpr int FIN = 128;     // IN_DIM + HID
constexpr int C   = 256;     // 4*HID
constexpr int HID = 64;
constexpr int EE  = E + N;   // edges + self loops
constexpr float NEG_SLOPE = 0.2f;

// ---------------------------------------------------------------------------
// Init workspace: zero accumulator, mx = -inf (bit pattern), denom = 0
// ---------------------------------------------------------------------------
__global__ void k_init(float* __restrict__ acc, float* __restrict__ mx,
                       float* __restrict__ den) {
    size_t i = (size_t)blockIdx.x * blockDim.x + threadIdx.x;
    if (i < (size_t)B * N * C) acc[i] = 0.0f;
    if (i < (size_t)B * N) {
        ((unsigned int*)mx)[i] = 0xFF800000u;  // -inf
        den[i] = 0.0f;
    }
}

// ---------------------------------------------------------------------------
// xw = concat(x,h) @ W^T via V_WMMA_F32_16X16X4_F32
// grid (N/16, B), 256 threads = 8 waves; wave w owns cols [32w, 32w+32)
// ---------------------------------------------------------------------------
__global__ __launch_bounds__(256) void k_gemm(const float* __restrict__ x,
                                              const float* __restrict__ h,
                                              const float* __restrict__ W,
                                              float* __restrict__ xw) {
    __shared__ float Als[16][132];  // +4 pad: row stride 132 mod 64 banks = 4
    const int b   = blockIdx.y;
    const int n0  = blockIdx.x * 16;
    const int tid = threadIdx.x;

    // Stage the 16x128 A tile (concat of input and h) into LDS, coalesced.
    #pragma unroll
    for (int j = 0; j < 8; ++j) {
        int e   = tid + 256 * j;
        int row = e >> 7;
        int col = e & 127;
        float v = (col < HID)
                      ? x[((size_t)b * N + n0 + row) * HID + col]
                      : h[((size_t)b * N + n0 + row) * HID + (col - HID)];
        Als[row][col] = v;
    }
    __syncthreads();

    const int lane = tid & 31;
    const int wid  = tid >> 5;
    const int lo   = lane & 15;
    const int hi   = lane >> 4;
    const int col0 = wid * 32 + lo;
    const int col1 = col0 + 16;

    v8f acc0 = {};
    v8f acc1 = {};
    #pragma unroll 8
    for (int k0 = 0; k0 < FIN; k0 += 4) {
        const int ka = k0 + 2 * hi;
        v2f a;  a.x  = Als[lo][ka];                  a.y  = Als[lo][ka + 1];
        v2f b0; b0.x = W[(size_t)col0 * FIN + ka];   b0.y = W[(size_t)col0 * FIN + ka + 1];
        v2f b1; b1.x = W[(size_t)col1 * FIN + ka];   b1.y = W[(size_t)col1 * FIN + ka + 1];
        acc0 = __builtin_amdgcn_wmma_f32_16x16x4_f32(false, a, false, b0,
                                                     (short)0, acc0, false, false);
        acc1 = __builtin_amdgcn_wmma_f32_16x16x4_f32(false, a, false, b1,
                                                     (short)0, acc1, false, false);
    }

    #pragma unroll
    for (int i = 0; i < 8; ++i) {
        int m = i + 8 * hi;  // C/D layout: VGPR i -> M=i (lanes 0-15), M=i+8 (16-31)
        size_t base = ((size_t)b * N + n0 + m) * C;
        xw[base + col0] = acc0[i];
        xw[base + col1] = acc1[i];
    }
}

// ---------------------------------------------------------------------------
// a_s[n] = dot(xw[n], att_src), a_d[n] = dot(xw[n], att_dst); wave per node
// ---------------------------------------------------------------------------
__global__ __launch_bounds__(256) void k_dots(const float* __restrict__ xw,
                                              const float* __restrict__ att_s,
                                              const float* __restrict__ att_d,
                                              float* __restrict__ as_,
                                              float* __restrict__ ad_) {
    int gwave = (blockIdx.x * blockDim.x + threadIdx.x) >> 5;
    int lane  = threadIdx.x & 31;
    if (gwave >= B * N) return;
    const float* row = &xw[(size_t)gwave * C];
    float ps = 0.0f, pd = 0.0f;
    #pragma unroll
    for (int j = 0; j < 8; ++j) {
        int cc = lane + 32 * j;
        float v = row[cc];
        ps += v * att_s[cc];
        pd += v * att_d[cc];
    }
    #pragma unroll
    for (int off = 16; off; off >>= 1) {
        ps += __shfl_xor(ps, off);
        pd += __shfl_xor(pd, off);
    }
    if (lane == 0) { as_[gwave] = ps; ad_[gwave] = pd; }
}

// Order-preserving float atomic max (works for mixed signs with -inf init)
__device__ inline void atomicMaxF(float* addr, float val) {
    if (val >= 0.0f) atomicMax((int*)addr, __float_as_int(val));
    else             atomicMin((unsigned int*)addr, (unsigned int)__float_as_int(val));
}

__device__ inline void edge_sd(const int* __restrict__ ei, int e, int& s, int& d) {
    if (e < E) { s = ei[e]; d = ei[E + e]; }
    else       { s = d = e - E; }  // self loop
}

// ---------------------------------------------------------------------------
// Pass 1: segment max of leaky_relu(a_s[src]+a_d[dst]) keyed by dst
// ---------------------------------------------------------------------------
__global__ void k_edge_max(const int* __restrict__ ei,
                           const float* __restrict__ as_,
                           const float* __restrict__ ad_,
                           float* __restrict__ mx) {
    int t = blockIdx.x * blockDim.x + threadIdx.x;
    if (t >= B * EE) return;
    int b = t / EE, e = t % EE, s, d;
    edge_sd(ei, e, s, d);
    float v = as_[b * N + s] + ad_[b * N + d];
    v = (v >= 0.0f) ? v : NEG_SLOPE * v;
    atomicMaxF(&mx[b * N + d], v);
}

// ---------------------------------------------------------------------------
// Pass 2: denom[dst] += exp(e - mx[dst])
// ---------------------------------------------------------------------------
__global__ void k_edge_sum(const int* __restrict__ ei,
                           const float* __restrict__ as_,
                           const float* __restrict__ ad_,
                           const float* __restrict__ mx,
                           float* __restrict__ den) {
    int t = blockIdx.x * blockDim.x + threadIdx.x;
    if (t >= B * EE) return;
    int b = t / EE, e = t % EE, s, d;
    edge_sd(ei, e, s, d);
    int bd = b * N + d;
    float v = as_[b * N + s] + ad_[bd];
    v = (v >= 0.0f) ? v : NEG_SLOPE * v;
    atomicAdd(&den[bd], expf(v - mx[bd]));
}

// ---------------------------------------------------------------------------
// Pass 3: acc[dst] += alpha * xw[src]   (wave per edge, 8 channels per lane)
// ---------------------------------------------------------------------------
__global__ __launch_bounds__(256) void k_scatter(const int* __restrict__ ei,
                                                 const float* __restrict__ as_,
                                                 const float* __restrict__ ad_,
                                                 const float* __restrict__ mx,
                                                 const float* __restrict__ den,
                                                 const float* __restrict__ xw,
                                                 float* __restrict__ acc) {
    int gw   = (blockIdx.x * blockDim.x + threadIdx.x) >> 5;
    int lane = threadIdx.x & 31;
    if (gw >= B * EE) return;
    int b = gw / EE, e = gw % EE, s, d;
    edge_sd(ei, e, s, d);
    int bs = b * N + s, bd = b * N + d;
    float v = as_[bs] + ad_[bd];
    v = (v >= 0.0f) ? v : NEG_SLOPE * v;
    float alpha = expf(v - mx[bd]) / (den[bd] + 1e-16f);
    const float* srow = &xw[(size_t)bs * C];
    float*       drow = &acc[(size_t)bd * C];
    #pragma unroll
    for (int j = 0; j < 8; ++j) {
        int cc = lane + 32 * j;
        atomicAdd(&drow[cc], alpha * srow[cc]);  // L2-resident float atomic
    }
}

// ---------------------------------------------------------------------------
// Fused bias + LSTM cell; writes h_next || c_next to d_out
// ---------------------------------------------------------------------------
__global__ void k_lstm(const float* __restrict__ acc,
                       const float* __restrict__ bias,
                       const float* __restrict__ c_cur,
                       float* __restrict__ out) {
    int t = blockIdx.x * blockDim.x + threadIdx.x;  // over B*N*HID
    if (t >= B * N * HID) return;
    int hch = t & (HID - 1);
    int bn  = t >> 6;
    const float* row = &acc[(size_t)bn * C];
    float ci = row[hch]       + bias[hch];
    float cf = row[hch + 64]  + bias[hch + 64];
    float co = row[hch + 128] + bias[hch + 128];
    float cg = row[hch + 192] + bias[hch + 192];
    float i_ = 1.0f / (1.0f + expf(-ci));
    float f_ = 1.0f / (1.0f + expf(-cf));
    float o_ = 1.0f / (1.0f + expf(-co));
    float g_ = tanhf(cg);
    float cn = f_ * c_cur[t] + i_ * g_;
    float hn = o_ * tanhf(cn);
    out[t] = hn;
    out[(size_t)B * N * HID + t] = cn;
}

// ---------------------------------------------------------------------------
extern "C" void kernel_launch(void* const* d_in, const int* in_sizes, int n_in,
                              void* d_out, int out_size, void* d_ws, size_t ws_size,
                              hipStream_t stream) {
    const float* x     = (const float*)d_in[0];
    const float* h     = (const float*)d_in[1];
    const float* ccur  = (const float*)d_in[2];
    const float* W     = (const float*)d_in[3];
    const float* att_s = (const float*)d_in[4];
    const float* att_d = (const float*)d_in[5];
    const float* bias  = (const float*)d_in[6];
    const int*   ei    = (const int*)d_in[7];

    float* ws  = (float*)d_ws;
    float* xw  = ws;                               // B*N*C
    float* acc = xw  + (size_t)B * N * C;          // B*N*C
    float* as_ = acc + (size_t)B * N * C;          // B*N
    float* ad_ = as_ + (size_t)B * N;              // B*N
    float* mx  = ad_ + (size_t)B * N;              // B*N
    float* den = mx  + (size_t)B * N;              // B*N
    float* out = (float*)d_out;

    k_init<<<((size_t)B * N * C + 255) / 256, 256, 0, stream>>>(acc, mx, den);
    k_gemm<<<dim3(N / 16, B), 256, 0, stream>>>(x, h, W, xw);
    k_dots<<<((size_t)B * N * 32 + 255) / 256, 256, 0, stream>>>(xw, att_s, att_d, as_, ad_);
    k_edge_max<<<(B * EE + 255) / 256, 256, 0, stream>>>(ei, as_, ad_, mx);
    k_edge_sum<<<(B * EE + 255) / 256, 256, 0, stream>>>(ei, as_, ad_, mx, den);
    k_scatter<<<((size_t)B * EE * 32 + 255) / 256, 256, 0, stream>>>(ei, as_, ad_, mx, den, xw, acc);
    k_lstm<<<(B * N * HID + 255) / 256, 256, 0, stream>>>(acc, bias, ccur, out);
}